// GRU_Network_31439160607292
// MI455X (gfx1250) — compile-verified
//
#include <hip/hip_runtime.h>

// GRU-style recurrence on MI455X (gfx1250, wave32).
// Phase 1: ih = x @ Wih^T + bih  -> fp32 WMMA GEMM (bandwidth-bound: 268 MB vs
//          17.2 GFLOP -> ~11.5us HBM floor at 23.3 TB/s), written into d_out.
// Phase 2: persistent 16-workgroup scan; Whh fragments register-resident,
//          h ping-pong in L2, device barrier per timestep (latency-bound).

#define AS1 __attribute__((address_space(1)))
#define AS3 __attribute__((address_space(3)))

namespace {

constexpr int kB = 128;
constexpr int kS = 1024;
constexpr int kD = 256;
constexpr int kPitch = 260;  // LDS pitch (floats): 260 % 64 == 4 -> conflict-free frag reads

typedef __attribute__((ext_vector_type(2))) float v2f;
typedef __attribute__((ext_vector_type(8))) float v8f;
typedef int v4i __attribute__((vector_size(16)));  // b128 payload type
typedef AS1 v4i* g_v4i_p;                          // global (AS1) b128 pointer
typedef AS3 v4i* l_v4i_p;                          // LDS (AS3) b128 pointer

#if __has_builtin(__builtin_amdgcn_global_load_async_to_lds_b128)
#define GRU_ASYNC_LDS 1
#else
#define GRU_ASYNC_LDS 0
#endif

__device__ __forceinline__ v8f wmma_f32_16x16x4(v2f a, v2f b, v8f c) {
  return __builtin_amdgcn_wmma_f32_16x16x4_f32(
      /*neg_a=*/false, a, /*neg_b=*/false, b,
      /*c_mod=*/(short)0, c, /*reuse_a=*/false, /*reuse_b=*/false);
}

__device__ __forceinline__ float fast_sigmoid(float x) {
  return 1.0f / (1.0f + __expf(-x));
}
__device__ __forceinline__ float fast_tanh(float x) {
  return 2.0f / (1.0f + __expf(-2.0f * x)) - 1.0f;  // NaN-safe at both extremes
}

// Stage a 16x256 fp32 tile (rows stride kD) into LDS at pitch kPitch.
// Uses the gfx1250 async-to-LDS path (ASYNCcnt + s_wait_asynccnt) when
// available; falls back to b128 load/ds_store otherwise.
__device__ __forceinline__ void stage16x256(const float* __restrict__ g,
                                            float* l, int tid) {
#pragma unroll
  for (int i = 0; i < 4; ++i) {
    const int c  = tid + 256 * i;  // float4 chunk 0..1023
    const int r  = c >> 6;         // row 0..15
    const int k4 = c & 63;         // float4 within row
    const float* gp = g + (size_t)r * kD + k4 * 4;
    float* lp = &l[r * kPitch + k4 * 4];
#if GRU_ASYNC_LDS
    __builtin_amdgcn_global_load_async_to_lds_b128(
        (g_v4i_p)(__UINTPTR_TYPE__)gp, (l_v4i_p)(__UINTPTR_TYPE__)lp,
        /*offset=*/0, /*cpol=*/0);
#else
    *reinterpret_cast<float4*>(lp) = *reinterpret_cast<const float4*>(gp);
#endif
  }
#if GRU_ASYNC_LDS
  asm volatile("s_wait_asynccnt 0x0" ::: "memory");
#endif
}

// ---------------------------------------------------------------------------
// Phase 1: ih[m, n] = sum_k x[m, k] * Wih[n, k] + bih[n]
// 256 threads / 8 waves per block; block = 16 rows x 128 cols; wave = one
// 16x16 tile, K=256 -> 64 fp32 WMMAs. A tile staged in LDS.
// ---------------------------------------------------------------------------
__global__ __launch_bounds__(256) void gru_ih_gemm(
    const float* __restrict__ x, const float* __restrict__ Wih,
    const float* __restrict__ bih, float* __restrict__ ih) {
  __shared__ __attribute__((aligned(16))) float lA[16 * kPitch];

  const int tid = threadIdx.x;
  const int m0  = blockIdx.x * 16;
  const int n0g = blockIdx.y * 128;

  stage16x256(x + (size_t)m0 * kD, lA, tid);
  __syncthreads();

  const int wave = tid >> 5;
  const int lane = tid & 31;
  const int l16  = lane & 15;
  const int hi   = lane >> 4;  // 0: K={0,1}, 1: K={2,3} per f32 A-frag layout
  const int n0   = n0g + wave * 16;

  const float* Ap = &lA[l16 * kPitch + hi * 2];
  const float* Bp = Wih + (size_t)(n0 + l16) * kD + hi * 2;  // B[k][n] = Wih[n][k]

  v8f acc = {};
#pragma unroll 8
  for (int kk = 0; kk < 64; ++kk) {
    const v2f a = *(const v2f*)(Ap + kk * 4);
    const v2f b = *(const v2f*)(Bp + kk * 4);
    acc = wmma_f32_16x16x4(a, b, acc);
  }

  const float bias = bih[n0 + l16];
#pragma unroll
  for (int r = 0; r < 8; ++r) {  // VGPR r: M = r (lanes 0-15) / r+8 (lanes 16-31)
    const int m = m0 + r + hi * 8;
    ih[(size_t)m * kD + (n0 + l16)] = acc[r] + bias;
  }
}

// ---------------------------------------------------------------------------
// Phase 2: persistent scan. 16 WGs x 8 waves = 128 waves; wave (wg=colTile,
// wave=rowTile) owns one 16x16 h tile for all 1024 steps. Whh fragments are
// loaded ONCE into 128 VGPRs (v2f breg[64]) -> per-step critical path is just
// A-loads (L2-resident h) + 64 WMMAs + gates. Monotonic-counter device
// barrier between steps (16 co-resident WGs -> deadlock-free).
// ---------------------------------------------------------------------------
__global__ __launch_bounds__(256) void gru_scan(
    float* __restrict__ seq,            // [B,S,D]: ih on entry, h on exit
    float* __restrict__ hT,             // [B,D]
    const float* __restrict__ Whh, const float* __restrict__ bhh,
    float* __restrict__ hbuf,           // 2 * B*D floats (buffer 0 = h0)
    unsigned* __restrict__ bar) {       // zeroed before launch
  __shared__ __attribute__((aligned(16))) float lW[16 * kPitch];

  const int tid = threadIdx.x;
  const int n0  = blockIdx.x * 16;  // 16 workgroups cover D = 256 columns

  stage16x256(Whh + (size_t)n0 * kD, lW, tid);
  __syncthreads();

  const int wave = tid >> 5;
  const int lane = tid & 31;
  const int l16  = lane & 15;
  const int hi   = lane >> 4;
  const int m0   = wave * 16;       // 8 waves cover B = 128 rows
  const int n    = n0 + l16;
  const float bias   = bhh[n];
  const unsigned nwg = gridDim.x;

  // Hoist the wave's Whh B-fragments into registers (loop-invariant).
  v2f breg[64];
  {
    const float* Bp = &lW[l16 * kPitch + hi * 2];
#pragma unroll
    for (int kk = 0; kk < 64; ++kk) breg[kk] = *(const v2f*)(Bp + kk * 4);
  }

  for (int t = 0; t < kS; ++t) {
    const float* hprev = hbuf + (t & 1) * (kB * kD);
    float* hnext       = hbuf + ((t + 1) & 1) * (kB * kD);

    // hp = h_prev @ Whh^T for this tile: K = 256 -> 64 fp32 WMMAs, B in regs.
    const float* Ap = hprev + (size_t)(m0 + l16) * kD + hi * 2;
    v8f acc = {};
#pragma unroll
    for (int kk = 0; kk < 64; ++kk) {
      const v2f a = *(const v2f*)(Ap + kk * 4);
      acc = wmma_f32_16x16x4(a, breg[kk], acc);
    }

    // Gates + state update (TRANS v_exp co-executes with the matrix pipe).
#pragma unroll
    for (int r = 0; r < 8; ++r) {
      const int m       = m0 + r + hi * 8;
      const size_t sidx = ((size_t)m * kS + t) * kD + n;
      const float it    = seq[sidx];          // i_t, consumed then overwritten
      const float hp    = acc[r] + bias;
      const float he    = hprev[m * kD + n];
      const float gz    = fast_sigmoid(it + hp);       // z == r in reference
      const float nt    = fast_tanh((it + gz) * hp);
      const float ht    = 1.0f - gz * nt + gz * he;    // reference precedence
      seq[sidx]          = ht;
      hnext[m * kD + n]  = ht;
      if (t == kS - 1) hT[m * kD + n] = ht;
    }

    // Prefetch next step's i_t tile while waiting at the barrier.
    if (t + 1 < kS) {
      __builtin_prefetch(&seq[((size_t)(m0 + l16) * kS + (t + 1)) * kD + n0],
                         0, 3);
    }

    // Device-wide step barrier: monotonic counter, no reset races.
    __threadfence();
    __syncthreads();
    if (tid == 0) {
      __hip_atomic_fetch_add(bar, 1u, __ATOMIC_ACQ_REL,
                             __HIP_MEMORY_SCOPE_AGENT);
      const unsigned target = (unsigned)(t + 1) * nwg;
      while (__hip_atomic_load(bar, __ATOMIC_ACQUIRE,
                               __HIP_MEMORY_SCOPE_AGENT) < target) {
        __builtin_amdgcn_s_sleep(1);  // 64-clk backoff: keep step latency low
      }
    }
    __syncthreads();
    __threadfence();
  }
}

}  // namespace

extern "C" void kernel_launch(void* const* d_in, const int* in_sizes, int n_in,
                              void* d_out, int out_size, void* d_ws,
                              size_t ws_size, hipStream_t stream) {
  const float* x   = (const float*)d_in[0];
  const float* h0  = (const float*)d_in[1];
  const float* Wih = (const float*)d_in[2];
  const float* bih = (const float*)d_in[3];
  const float* Whh = (const float*)d_in[4];
  const float* bhh = (const float*)d_in[5];

  float* seq = (float*)d_out;                    // [B,S,D]
  float* hT  = seq + (size_t)kB * kS * kD;       // [B,D]

  unsigned* bar = (unsigned*)d_ws;               // barrier counter @ offset 0
  float* hbuf   = (float*)((char*)d_ws + 256);   // 2 x B*D fp32 ping-pong

  const size_t hbytes = (size_t)kB * kD * sizeof(float);
  (void)hipMemsetAsync(d_ws, 0, 256 + 2 * hbytes, stream);                 // bar=0
  (void)hipMemcpyAsync(hbuf, h0, hbytes, hipMemcpyDeviceToDevice, stream); // h_prev

  // Phase 1: ih -> seq (in-place reuse of hidden_seq storage).
  gru_ih_gemm<<<dim3((kB * kS) / 16, kD / 128), 256, 0, stream>>>(x, Wih, bih,
                                                                  seq);
  // Phase 2: persistent scan, 16 co-resident workgroups.
  gru_scan<<<dim3(16), 256, 0, stream>>>(seq, hT, Whh, bhh, hbuf, bar);
}